// GridSpatialIntegral_38989713113240
// MI455X (gfx1250) — compile-verified
//
#include <hip/hip_runtime.h>
#include <stdint.h>

// GridSpatialIntegral on MI455X (gfx1250):
//   out[:,0] = cumsum(in[:,0], axis=x)  -- WMMA f32 triangular-matmul scan via LDS
//   out[:,1] = cumsum(in[:,1], axis=y)  -- coalesced serial column scan
// Memory-bound: ~268 MB traffic -> ~11.5 us floor @ 23.3 TB/s. WMMA adds only
// ~0.54 GFLOP f32 (32 FLOP/elem on ch0), far below the matrix-pipe ceiling.

typedef __attribute__((ext_vector_type(2))) float v2f;
typedef __attribute__((ext_vector_type(4))) float v4f;
typedef __attribute__((ext_vector_type(8))) float v8f;

#define PITCH 516  // 512 + 4 pad floats: lane stride = 4 banks -> conflict-free

__global__ __launch_bounds__(256) void gsi_scan_x_wmma(const float* __restrict__ in,
                                                       float* __restrict__ out) {
  __shared__ float sm[16 * PITCH];   // 16 x 512 panel, padded
  __shared__ float sums[16 * 32];    // per-row chunk totals
  __shared__ float carry[16 * 32];   // per-row exclusive chunk prefix

  const int t     = threadIdx.x;
  const int p     = blockIdx.x;      // 2048 panels: 64 b * 32 row-panels
  const int b     = p >> 5;
  const int panel = p & 31;
  // channel 0 of batch b, rows [panel*16, panel*16+16): contiguous 8192 floats
  const size_t base = ((size_t)(b * 2) * 512 + (size_t)panel * 16) * 512;

  // ---- Stage 1: async-copy panel global -> LDS (per-lane 16B transfers) ----
  #pragma unroll
  for (int i = 0; i < 8; ++i) {
    const int e = (t + 256 * i) * 4;           // float index in panel, %4==0
    const int m = e >> 9;                      // row in panel
    const int x = e & 511;                     // col
    unsigned lds_off = (unsigned)(uintptr_t)&sm[m * PITCH + x];
    uint64_t ga = (uint64_t)(uintptr_t)(in + base + e);
    asm volatile("global_load_async_to_lds_b128 %0, %1, off"
                 :: "v"(lds_off), "v"(ga) : "memory");
  }
  asm volatile("s_wait_asynccnt 0" ::: "memory");
  __syncthreads();

  // ---- Stage 2: per-chunk inclusive scan as D = T x L (L upper-tri ones) ----
  const int lane = t & 31;
  const int wave = t >> 5;
  const int mrow = lane & 15;   // A-fragment row / D-fragment column N
  const int half = lane >> 4;   // K-halves for A/B, M-halves for C/D

  #pragma unroll
  for (int cc = 0; cc < 4; ++cc) {
    const int c = wave * 4 + cc;    // chunk id 0..31 (disjoint across waves)
    const int cbase = c * 16;
    v8f acc = {};
    #pragma unroll
    for (int s = 0; s < 4; ++s) {   // chain 4x K=4 -> 16x16x16 f32 product
      const int k0 = 4 * s + 2 * half;
      // A (16x4 f32): lane holds T[mrow][k0], T[mrow][k0+1]  (ds_load_b64)
      const v2f a = *(const v2f*)&sm[mrow * PITCH + cbase + k0];
      // B (4x16): L[k][n] = (k <= n), n = mrow; constant, built in registers
      v2f bv;
      bv.x = (k0     <= mrow) ? 1.0f : 0.0f;
      bv.y = (k0 + 1 <= mrow) ? 1.0f : 0.0f;
      acc = __builtin_amdgcn_wmma_f32_16x16x4_f32(false, a, false, bv,
                                                  (short)0, acc, false, false);
    }
    // D layout: VGPR r, lanes 0-15 -> (M=r, N=lane); lanes 16-31 -> (M=r+8)
    #pragma unroll
    for (int r = 0; r < 8; ++r)
      sm[(r + 8 * half) * PITCH + cbase + mrow] = acc[r];   // scanned in place
    if (mrow == 15) {               // column 15 = chunk total per row
      #pragma unroll
      for (int r = 0; r < 8; ++r)
        sums[(r + 8 * half) * 32 + c] = acc[r];
    }
  }
  __syncthreads();

  // ---- Exclusive scan of 32 chunk totals per row (16 rows, 1 thread each) ----
  if (t < 16) {
    float run = 0.0f;
    #pragma unroll
    for (int c2 = 0; c2 < 32; ++c2) {
      carry[t * 32 + c2] = run;
      run += sums[t * 32 + c2];
    }
  }
  __syncthreads();

  // ---- Stage 3: add carries, coalesced float4 stores ----
  #pragma unroll
  for (int i = 0; i < 8; ++i) {
    const int e = (t + 256 * i) * 4;
    const int m = e >> 9;
    const int x = e & 511;
    const float cy = carry[m * 32 + (x >> 4)];
    const v4f sv = *(const v4f*)&sm[m * PITCH + x];   // 16B aligned (PITCH even, x%4==0)
    v4f v;
    v.x = sv.x + cy; v.y = sv.y + cy; v.z = sv.z + cy; v.w = sv.w + cy;
    *(v4f*)(out + base + e) = v;
  }
}

__global__ __launch_bounds__(256) void gsi_scan_y(const float* __restrict__ in,
                                                  float* __restrict__ out) {
  const int t = blockIdx.x * 256 + threadIdx.x;   // 32768 columns total
  const int b = t >> 9;
  const int x = t & 511;
  const size_t base = ((size_t)(b * 2 + 1) * 512) * 512 + x;  // channel 1
  float acc = 0.0f;
  #pragma unroll 8
  for (int y = 0; y < 512; ++y) {
    acc += in[base + (size_t)y * 512];   // consecutive lanes -> consecutive addrs
    out[base + (size_t)y * 512] = acc;
  }
}

extern "C" void kernel_launch(void* const* d_in, const int* in_sizes, int n_in,
                              void* d_out, int out_size, void* d_ws, size_t ws_size,
                              hipStream_t stream) {
  (void)in_sizes; (void)n_in; (void)out_size; (void)d_ws; (void)ws_size;
  const float* in = (const float*)d_in[0];
  float* out = (float*)d_out;
  // Channel 0: 64 batches * 32 row-panels = 2048 blocks of 8 waves
  gsi_scan_x_wmma<<<2048, 256, 0, stream>>>(in, out);
  // Channel 1: 32768 columns / 256 = 128 blocks
  gsi_scan_y<<<128, 256, 0, stream>>>(in, out);
}